// TimeGSMKernel_14448269984294
// MI455X (gfx1250) — compile-verified
//
#include <hip/hip_runtime.h>
#include <math.h>

// Problem constants (fixed by the reference):
#define B_   4
#define L_   1024
#define TD_  16
#define TWO_PI_F 6.28318530717958647692f

typedef __attribute__((ext_vector_type(2))) float v2f;
typedef __attribute__((ext_vector_type(8))) float v8f;

// ---------------------------------------------------------------------------
// One 16-row tile of the 2-layer FFN, second layer done with WMMA f32 16x16x4.
//   layer1: h[m][d] = relu(x[m]*W1[d] + b1[d])          (x is scalar per row)
//   layer2: o = relu(h @ W2 + b2)                        (16x16 @ 16x16 GEMM)
// A-matrix layout (16x4 f32, ISA 7.12.2): lane L holds row M=L%16,
//   VGPR v holds K = 4t + v + 2*(L>=16).  B uses the mirrored K layout with
//   N = L%16.  C/D: VGPR r, lane L -> (M = r + 8*(L>=16), N = L%16).
// ---------------------------------------------------------------------------
__device__ __forceinline__ v8f ffn16_tile(float xm, int hi, int n,
                                          const float* __restrict__ W1,
                                          const float* __restrict__ b1,
                                          const float* __restrict__ W2,
                                          const float* __restrict__ b2)
{
    v8f c;
    float bias = b2[n];
#pragma unroll
    for (int r = 0; r < 8; ++r) c[r] = bias;

#pragma unroll
    for (int t = 0; t < 4; ++t) {
        int k0 = 4 * t + 2 * hi;
        v2f a, bm;
        a.x  = fmaxf(fmaf(xm, W1[k0],     b1[k0]),     0.0f);
        a.y  = fmaxf(fmaf(xm, W1[k0 + 1], b1[k0 + 1]), 0.0f);
        bm.x = W2[(k0)     * 16 + n];
        bm.y = W2[(k0 + 1) * 16 + n];
        // D = A*B + C   (v_wmma_f32_16x16x4_f32)
        c = __builtin_amdgcn_wmma_f32_16x16x4_f32(false, a, false, bm,
                                                  (short)0, c, false, false);
    }
#pragma unroll
    for (int r = 0; r < 8; ++r) c[r] = fmaxf(c[r], 0.0f);
    return c;
}

// ---------------------------------------------------------------------------
// Stage 1: per-row features.  feat[row*16+d] = {cos(qpq), sin(qpq), s^2, a}
//   qpq = 2pi * ffn_p(x) * x,  s = ffn_s(x)+1e-6,  a = ascale * ffn_b(x)*sqrt(s)
// ascale = sqrt(2) on the q side (folds the sqrt(2) of local_enc), 1 on k side.
// Block = 256 threads = 8 waves, each wave owns a 16-row tile.
// ---------------------------------------------------------------------------
__global__ void __launch_bounds__(256)
feat_kernel(const float* __restrict__ x, float4* __restrict__ feat,
            const float* __restrict__ pW1, const float* __restrict__ pb1,
            const float* __restrict__ pW2, const float* __restrict__ pb2,
            const float* __restrict__ sW1, const float* __restrict__ sb1,
            const float* __restrict__ sW2, const float* __restrict__ sb2,
            const float* __restrict__ bW1, const float* __restrict__ bb1,
            const float* __restrict__ bW2, const float* __restrict__ bb2,
            float ascale)
{
    int lane = threadIdx.x & 31;
    int wave = threadIdx.x >> 5;
    int hi   = lane >> 4;     // which half-wave (selects K / M groups)
    int n    = lane & 15;     // A row index & B/C column index
    int r0   = (blockIdx.x * 8 + wave) * 16;

    float xm = x[r0 + n];     // row n of this tile (lanes n and n+16 both hold it)

    v8f cp = ffn16_tile(xm, hi, n, pW1, pb1, pW2, pb2);
    v8f cs = ffn16_tile(xm, hi, n, sW1, sb1, sW2, sb2);
    v8f cb = ffn16_tile(xm, hi, n, bW1, bb1, bW2, bb2);

#pragma unroll
    for (int r = 0; r < 8; ++r) {
        int   m  = r + 8 * hi;            // output row owned by this VGPR
        float xr = __shfl(xm, m, 32);     // x of row m (lane m holds it)
        float qpq = TWO_PI_F * cp[r] * xr;
        float s   = cs[r] + 1e-6f;
        float4 f;
        f.x = __cosf(qpq);
        f.y = __sinf(qpq);
        f.z = s * s;
        f.w = ascale * cb[r] * __builtin_sqrtf(s);
        feat[(r0 + m) * 16 + n] = f;      // d = n
    }
}

// ---------------------------------------------------------------------------
// Stage 2: pair reduction.  Each thread owns one k-column j; q-tile features
// (16 rows x 16 d) staged in LDS (broadcast reads, conflict-free).
//   out[i,j] = sum_d aq*ak * rsq(denom) * exp(-(qi-kj)^2 / denom)
//                    * (cq*ck + snq*snk),   denom = sq^2 + sk^2
// ---------------------------------------------------------------------------
__global__ void __launch_bounds__(256)
pair_kernel(const float* __restrict__ q, const float* __restrict__ k,
            const float4* __restrict__ fq, const float4* __restrict__ fk,
            float* __restrict__ out)
{
    __shared__ float4 qf[16 * 16];
    __shared__ float  qv[16];

    int b  = blockIdx.z;
    int i0 = blockIdx.y * 16;
    int j  = blockIdx.x * 256 + threadIdx.x;
    int t  = threadIdx.x;

    qf[t] = fq[(size_t)(b * L_ + i0 + (t >> 4)) * 16 + (t & 15)];
    if (t < 16) qv[t] = q[b * L_ + i0 + t];
    __syncthreads();

    float  kj = k[b * L_ + j];
    float4 kk[16];
#pragma unroll
    for (int d = 0; d < 16; ++d) kk[d] = fk[(size_t)(b * L_ + j) * 16 + d];

#pragma unroll 2
    for (int i = 0; i < 16; ++i) {
        float diff = qv[i] - kj;
        float nd2  = -diff * diff;
        float acc  = 0.0f;
#pragma unroll
        for (int d = 0; d < 16; ++d) {
            float4 f   = qf[i * 16 + d];
            float denom = f.z + kk[d].z;                  // sq^2 + sk^2
            float rd    = __builtin_amdgcn_rsqf(denom);   // v_rsq_f32
            float invd  = rd * rd;                        // 1/denom
            float e     = __expf(nd2 * invd);             // v_exp_f32
            float cosv  = fmaf(f.x, kk[d].x, f.y * kk[d].y);
            acc = fmaf(f.w * kk[d].w, rd * e * cosv, acc);
        }
        out[(size_t)(b * L_ + i0 + i) * L_ + j] = acc;
    }
}

// ---------------------------------------------------------------------------
extern "C" void kernel_launch(void* const* d_in, const int* in_sizes, int n_in,
                              void* d_out, int out_size, void* d_ws, size_t ws_size,
                              hipStream_t stream)
{
    const float* q   = (const float*)d_in[0];
    const float* kx  = (const float*)d_in[1];
    const float* pW1 = (const float*)d_in[2];
    const float* pb1 = (const float*)d_in[3];
    const float* pW2 = (const float*)d_in[4];
    const float* pb2 = (const float*)d_in[5];
    const float* sW1 = (const float*)d_in[6];
    const float* sb1 = (const float*)d_in[7];
    const float* sW2 = (const float*)d_in[8];
    const float* sb2 = (const float*)d_in[9];
    const float* bW1 = (const float*)d_in[10];
    const float* bb1 = (const float*)d_in[11];
    const float* bW2 = (const float*)d_in[12];
    const float* bb2 = (const float*)d_in[13];
    float* out = (float*)d_out;

    // Workspace: q features then k features, each B*L*16 float4 (1 MB each).
    float4* fq = (float4*)d_ws;
    float4* fk = fq + (size_t)B_ * L_ * TD_;

    const float SQRT2 = 1.41421356237309504880f;
    // 4096 rows / 128 rows-per-block = 32 blocks
    feat_kernel<<<32, 256, 0, stream>>>(q,  fq, pW1, pb1, pW2, pb2,
                                        sW1, sb1, sW2, sb2,
                                        bW1, bb1, bW2, bb2, SQRT2);
    feat_kernel<<<32, 256, 0, stream>>>(kx, fk, pW1, pb1, pW2, pb2,
                                        sW1, sb1, sW2, sb2,
                                        bW1, bb1, bW2, bb2, 1.0f);

    pair_kernel<<<dim3(L_ / 256, L_ / 16, B_), 256, 0, stream>>>(q, kx, fq, fk, out);
}